// ContinuousNormalizingFlowODE_59167469470113
// MI455X (gfx1250) — compile-verified
//
#include <hip/hip_runtime.h>
#include <hip/hip_bf16.h>

// ---------------------------------------------------------------------------
// Continuous Normalizing Flow ODE, fused Euler integrator for MI455X (gfx1250)
//
//   B=512 samples, D=64 state, H=512 hidden, T=8 times (7 Euler steps)
//
// Exact trace identity:  tr(J) = s2^T (W2 .* C^T)^T s1  with C = W3 @ W1[:64,:]
// -> prep kernel builds M[j,k] = W2[j,k] * C[k,j]  (time/sample independent)
// -> per step, trace costs one batched GEMM  V = S1 @ M, tr_b = sum_k s2*V
//
// One persistent workgroup per 32 samples, all 7 steps fused; f32 WMMA
// (v_wmma_f32_16x16x4_f32) for all four GEMMs; state in 320KB CDNA5 LDS;
// activations use hardware V_TANH_F32 (TRANS class, co-executes with WMMA).
// ---------------------------------------------------------------------------

#define BB  512
#define DD  64
#define HH  512
#define TT  8
#define MT  32          // samples per workgroup
#define XP  68          // LDS pitch for X   (68 % 64 == 4 -> conflict-free A reads)
#define HP  516         // LDS pitch for H1/H2

typedef float v2f __attribute__((ext_vector_type(2)));
typedef float v8f __attribute__((ext_vector_type(8)));

__device__ __forceinline__ v8f wmma_f32(v2f a, v2f b, v8f c) {
  // D = A(16x4) * B(4x16) + C(16x16), all f32.  8-arg form:
  // (neg_a, A, neg_b, B, c_mod, C, reuse_a, reuse_b)
  return __builtin_amdgcn_wmma_f32_16x16x4_f32(false, a, false, b,
                                               (short)0, c, false, false);
}

// Branchless tanh: prefer hardware V_TANH_F32 (gfx1250 TRANS op), then an
// exp2+rcp expansion (v_exp_f32 + v_rcp_f32, no EXEC divergence), then libm.
__device__ __forceinline__ float fast_tanh(float x) {
#if __has_builtin(__builtin_amdgcn_tanhf)
  return __builtin_amdgcn_tanhf(x);
#elif __has_builtin(__builtin_amdgcn_exp2f) && __has_builtin(__builtin_amdgcn_rcpf)
  // tanh(x) = 1 - 2/(exp(2x)+1);  exp(2x) = exp2(x * 2*log2(e))
  float e = __builtin_amdgcn_exp2f(x * 2.8853900817779268f);
  return 1.0f - 2.0f * __builtin_amdgcn_rcpf(e + 1.0f);
#else
  return tanhf(x);
#endif
}

// A fragment (16x4) from an LDS tile, row-major with pitch `pitch`.
// lanes 0-15: M=lane, K = k0+{0,1};  lanes 16-31: M=lane-16, K = k0+{2,3}.
__device__ __forceinline__ v2f frag_a(const float* S, int pitch, int m0, int k0,
                                      int lane_lo, int half) {
  const float* p = S + (m0 + lane_lo) * pitch + k0 + 2 * half;
  v2f a; a.x = p[0]; a.y = p[1];
  return a;
}

// B fragment (4x16) from a global row-major matrix with leading dim `ldw`.
// V0: lanes 0-15 = row k0, lanes 16-31 = row k0+2; V1: rows k0+1 / k0+3.
__device__ __forceinline__ v2f frag_b(const float* __restrict__ W, int ldw,
                                      int k0, int n0, int lane_lo, int half) {
  v2f b;
  b.x = W[(k0 + 2 * half + 0) * ldw + n0 + lane_lo];
  b.y = W[(k0 + 2 * half + 1) * ldw + n0 + lane_lo];
  return b;
}

// --------------------------- prep: M = W2 .* (W3 @ W1a)^T -------------------
__global__ __launch_bounds__(256) void cnf_prep(const float* __restrict__ W1,
                                                const float* __restrict__ W2,
                                                const float* __restrict__ W3,
                                                float* __restrict__ M) {
  int idx = blockIdx.x * blockDim.x + threadIdx.x;   // 512*512 elements
  int j = idx >> 9;          // row of W2  (hidden-1 index)
  int k = idx & 511;         // col of W2  (hidden-2 index)
  float c = 0.f;
  #pragma unroll 8
  for (int m = 0; m < DD; ++m)
    c += W3[k * DD + m] * W1[m * HH + j];            // C[k,j] = (W3 @ W1a)[k,j]
  M[j * HH + k] = W2[j * HH + k] * c;
}

// --------------------------- fused 7-step Euler integrator ------------------
__global__ __launch_bounds__(256) void cnf_main(const float* __restrict__ y0,
                                                const float* __restrict__ ts,
                                                const float* __restrict__ W1,
                                                const float* __restrict__ b1,
                                                const float* __restrict__ W2,
                                                const float* __restrict__ b2,
                                                const float* __restrict__ W3,
                                                const float* __restrict__ b3,
                                                const float* __restrict__ Mtr,
                                                float* __restrict__ out) {
  extern __shared__ float smem[];
  float* Xs   = smem;                    // [MT][XP]
  float* H1s  = Xs  + MT * XP;           // [MT][HP]   tanh layer-1 activations
  float* H2s  = H1s + MT * HP;           // [MT][HP]   tanh layer-2 activations
  float* zacc = H2s + MT * HP;           // [MT]       0.5*||f||^2 integral
  float* lacc = zacc + MT;               // [MT]       trace integral

  const int tid     = threadIdx.x;
  const int wave    = tid >> 5;
  const int lane    = tid & 31;
  const int lane_lo = lane & 15;
  const int half    = lane >> 4;
  const int s_base  = blockIdx.x * MT;   // first sample of this workgroup

  // ---- load initial state, zero integrals --------------------------------
  for (int i = tid; i < MT * DD; i += 256) {
    int m = i >> 6, c = i & 63;
    Xs[m * XP + c] = y0[(s_base + m) * DD + c];
  }
  if (tid < MT) { zacc[tid] = 0.f; lacc[tid] = 0.f; }

  v8f acc[2][4];    // 2 M-tiles x 4 N-tiles of 16x16 per wave

  for (int step = 0; step < TT - 1; ++step) {
    const float t0 = ts[step];
    const float dt = ts[step + 1] - t0;
    __syncthreads();                               // Xs/H1s stable

    // ================= phase 1: A1 = [X,t] @ W1 + b1 ; H1 = tanh ==========
    #pragma unroll
    for (int mt = 0; mt < 2; ++mt)
      #pragma unroll
      for (int nt = 0; nt < 4; ++nt) acc[mt][nt] = (v8f){};
    for (int k0 = 0; k0 < DD; k0 += 4) {
      v2f a0 = frag_a(Xs, XP, 0,  k0, lane_lo, half);
      v2f a1 = frag_a(Xs, XP, 16, k0, lane_lo, half);
      #pragma unroll
      for (int nt = 0; nt < 4; ++nt) {
        int n0 = (wave * 4 + nt) * 16;
        v2f b = frag_b(W1, HH, k0, n0, lane_lo, half);
        acc[0][nt] = wmma_f32(a0, b, acc[0][nt]);
        acc[1][nt] = wmma_f32(a1, b, acc[1][nt]);
      }
    }
    #pragma unroll
    for (int nt = 0; nt < 4; ++nt) {
      int col = (wave * 4 + nt) * 16 + lane_lo;
      float add = t0 * W1[DD * HH + col] + b1[col];   // t-column + bias
      #pragma unroll
      for (int mt = 0; mt < 2; ++mt)
        #pragma unroll
        for (int v = 0; v < 8; ++v) {
          int m = mt * 16 + v + 8 * half;
          H1s[m * HP + col] = fast_tanh(acc[mt][nt][v] + add);
        }
    }
    __syncthreads();

    // ================= phase 2: H2 = tanh(H1 @ W2 + b2) ====================
    #pragma unroll
    for (int mt = 0; mt < 2; ++mt)
      #pragma unroll
      for (int nt = 0; nt < 4; ++nt) acc[mt][nt] = (v8f){};
    for (int k0 = 0; k0 < HH; k0 += 4) {
      v2f a0 = frag_a(H1s, HP, 0,  k0, lane_lo, half);
      v2f a1 = frag_a(H1s, HP, 16, k0, lane_lo, half);
      #pragma unroll
      for (int nt = 0; nt < 4; ++nt) {
        int n0 = (wave * 4 + nt) * 16;
        v2f b = frag_b(W2, HH, k0, n0, lane_lo, half);
        acc[0][nt] = wmma_f32(a0, b, acc[0][nt]);
        acc[1][nt] = wmma_f32(a1, b, acc[1][nt]);
      }
    }
    #pragma unroll
    for (int nt = 0; nt < 4; ++nt) {
      int col = (wave * 4 + nt) * 16 + lane_lo;
      float add = b2[col];
      #pragma unroll
      for (int mt = 0; mt < 2; ++mt)
        #pragma unroll
        for (int v = 0; v < 8; ++v) {
          int m = mt * 16 + v + 8 * half;
          H2s[m * HP + col] = fast_tanh(acc[mt][nt][v] + add);
        }
    }
    __syncthreads();

    // ================= phase 3a: dx = H2 @ W3 + b3 ; update x, z ===========
    {
      int m0 = (wave >> 2) * 16;                  // one 16x16 tile per wave
      int n0 = (wave & 3) * 16;
      v8f a3 = (v8f){};
      for (int k0 = 0; k0 < HH; k0 += 4) {
        v2f a = frag_a(H2s, HP, m0, k0, lane_lo, half);
        v2f b = frag_b(W3, DD, k0, n0, lane_lo, half);
        a3 = wmma_f32(a, b, a3);
      }
      int col = n0 + lane_lo;
      float bb = b3[col];
      #pragma unroll
      for (int v = 0; v < 8; ++v) {
        int m = m0 + v + 8 * half;
        float dx = a3[v] + bb;
        atomicAdd(&zacc[m], dt * 0.5f * dx * dx);     // ds_add_f32
        Xs[m * XP + col] += dt * dx;                  // Euler update
      }
    }

    // ======== phase 3b: exact trace  tr_b = s2^T (M^T s1) ==================
    #pragma unroll
    for (int mt = 0; mt < 2; ++mt)
      #pragma unroll
      for (int nt = 0; nt < 4; ++nt) acc[mt][nt] = (v8f){};
    for (int k0 = 0; k0 < HH; k0 += 4) {
      v2f a0 = frag_a(H1s, HP, 0,  k0, lane_lo, half);
      v2f a1 = frag_a(H1s, HP, 16, k0, lane_lo, half);
      a0.x = 1.f - a0.x * a0.x;  a0.y = 1.f - a0.y * a0.y;   // s1 = 1 - h1^2
      a1.x = 1.f - a1.x * a1.x;  a1.y = 1.f - a1.y * a1.y;
      #pragma unroll
      for (int nt = 0; nt < 4; ++nt) {
        int n0 = (wave * 4 + nt) * 16;
        v2f b = frag_b(Mtr, HH, k0, n0, lane_lo, half);
        acc[0][nt] = wmma_f32(a0, b, acc[0][nt]);
        acc[1][nt] = wmma_f32(a1, b, acc[1][nt]);
      }
    }
    #pragma unroll
    for (int mt = 0; mt < 2; ++mt)
      #pragma unroll
      for (int v = 0; v < 8; ++v) {
        int m = mt * 16 + v + 8 * half;
        float tl = 0.f;
        #pragma unroll
        for (int nt = 0; nt < 4; ++nt) {
          int col = (wave * 4 + nt) * 16 + lane_lo;
          float h2 = H2s[m * HP + col];
          tl += (1.f - h2 * h2) * acc[mt][nt][v];     // s2 .* V
        }
        atomicAdd(&lacc[m], dt * tl);
      }
  }

  // ---- write (xf, zf, lf) concatenated -----------------------------------
  __syncthreads();
  for (int i = tid; i < MT * DD; i += 256) {
    int m = i >> 6, c = i & 63;
    out[(s_base + m) * DD + c] = Xs[m * XP + c];
  }
  if (tid < MT) {
    out[BB * DD + s_base + tid]      = zacc[tid];
    out[BB * DD + BB + s_base + tid] = lacc[tid];
  }
}

// ---------------------------------------------------------------------------
extern "C" void kernel_launch(void* const* d_in, const int* in_sizes, int n_in,
                              void* d_out, int out_size, void* d_ws, size_t ws_size,
                              hipStream_t stream) {
  const float* y0 = (const float*)d_in[0];
  const float* ts = (const float*)d_in[1];
  const float* W1 = (const float*)d_in[2];
  const float* b1 = (const float*)d_in[3];
  const float* W2 = (const float*)d_in[4];
  const float* b2 = (const float*)d_in[5];
  const float* W3 = (const float*)d_in[6];
  const float* b3 = (const float*)d_in[7];
  float* out = (float*)d_out;
  float* Mtr = (float*)d_ws;            // [512,512] trace bilinear-form matrix

  cnf_prep<<<(HH * HH) / 256, 256, 0, stream>>>(W1, W2, W3, Mtr);

  size_t shmem = (size_t)(MT * XP + 2 * MT * HP + 2 * MT) * sizeof(float);
  cnf_main<<<BB / MT, 256, shmem, stream>>>(y0, ts, W1, b1, W2, b2, W3, b3,
                                            Mtr, out);
}